// DVQuantumLinear_50036368998717
// MI455X (gfx1250) — compile-verified
//
#include <hip/hip_runtime.h>
#include <math.h>

typedef __attribute__((ext_vector_type(2))) float v2f;
typedef __attribute__((ext_vector_type(8))) float v8f;

#define NQ 4
#define DIM 16

// ---------------------------------------------------------------------------
// Setup kernel (1 wave): build the two folded 16x16 quadratic-form matrices
//   A_j = Ur^T diag(wt_j) Ur + Ui^T diag(wt_j) Ui      (j = 0,1)  [symmetric]
// where U is the fixed shared-gate unitary and wt_j(m) = sum_q head_w[j,q]*sign_q(m).
// Result -> ws[0..511] (A0 row-major, then A1 row-major).
// ---------------------------------------------------------------------------
__global__ __launch_bounds__(32) void dvq_setup(const float* __restrict__ weights,
                                                const float* __restrict__ head_w,
                                                float* __restrict__ ws) {
  __shared__ float Ur[DIM][DIM];
  __shared__ float Ui[DIM][DIM];
  const int lane = threadIdx.x;

  if (lane < DIM) {
    // propagate basis state e_lane through the shared circuit
    float re[DIM], im[DIM];
#pragma unroll
    for (int k = 0; k < DIM; ++k) { re[k] = 0.f; im[k] = 0.f; }
    re[lane] = 1.0f;

#pragma unroll
    for (int rep = 0; rep < 2; ++rep) {
#pragma unroll
      for (int i = 0; i < NQ; ++i) {
        const int msk = 1 << (3 - i);           // wire i -> bit (3-i)
        const float th_z = weights[i * 4 + 0];  // weights[0,i,0]
        const float th_y = weights[i * 4 + 1];  // weights[0,i,1]
        float shz, chz; __sincosf(0.5f * th_z, &shz, &chz);
        // RZ: bit==0 -> *(chz - i*shz) ; bit==1 -> *(chz + i*shz)
#pragma unroll
        for (int k = 0; k < DIM; ++k) {
          const float sgn = (k & msk) ? 1.0f : -1.0f;
          const float r = re[k], q = im[k];
          re[k] = r * chz - sgn * q * shz;
          im[k] = q * chz + sgn * r * shz;
        }
        float shy, chy; __sincosf(0.5f * th_y, &shy, &chy);
        // RY: [[c,-s],[s,c]] on the (bit=0, bit=1) pair
#pragma unroll
        for (int k = 0; k < DIM; ++k) {
          if (k & msk) continue;
          const int k1 = k | msk;
          const float r0 = re[k], q0 = im[k], r1 = re[k1], q1 = im[k1];
          re[k]  = chy * r0 - shy * r1;  im[k]  = chy * q0 - shy * q1;
          re[k1] = shy * r0 + chy * r1;  im[k1] = shy * q0 + chy * q1;
        }
      }
      // CNOT(control=wire0 (bit3), target=wire1 (bit2)): swap 8+t <-> 12+t
#pragma unroll
      for (int t = 0; t < 4; ++t) {
        const float tr = re[8 + t], ti = im[8 + t];
        re[8 + t] = re[12 + t]; im[8 + t] = im[12 + t];
        re[12 + t] = tr;        im[12 + t] = ti;
      }
    }
#pragma unroll
    for (int m = 0; m < DIM; ++m) { Ur[m][lane] = re[m]; Ui[m][lane] = im[m]; }
  }
  __syncthreads();

  // A_j[k][n] = sum_m wt_j[m] * (Ur[m][k]*Ur[m][n] + Ui[m][k]*Ui[m][n])
  for (int e = lane; e < 2 * DIM * DIM; e += 32) {
    const int j = e >> 8, k = (e >> 4) & 15, n = e & 15;
    float accv = 0.f;
#pragma unroll
    for (int m = 0; m < DIM; ++m) {
      float wt = 0.f;
#pragma unroll
      for (int q = 0; q < NQ; ++q) {
        const float sgn = ((m >> (3 - q)) & 1) ? -1.0f : 1.0f;
        wt += head_w[j * NQ + q] * sgn;
      }
      accv += wt * (Ur[m][k] * Ur[m][n] + Ui[m][k] * Ui[m][n]);
    }
    ws[e] = accv;
  }
}

// ---------------------------------------------------------------------------
// Main kernel: one wave per 16-sample tile (grid-strided).
// A_j symmetric  =>  W_j = A_j * Psi^T (= (Psi*A_j)^T) via 4 chained
// V_WMMA_F32_16X16X4_F32 per matrix. Lane lo owns D-column (= sample) lo, so
// the quadratic-form dot is lane-local + one shfl_xor(16).
// psi is NEVER materialized as an indexed array: all operands are built from
// the rank-1 factors ab[b0b1], cd[b2b3] with 4 scalar cndmask selects total,
// avoiding dynamic-extract compare ladders.
// ---------------------------------------------------------------------------
__global__ __launch_bounds__(256) void dvq_main(const float* __restrict__ x,
                                                const float* __restrict__ ws,
                                                const float* __restrict__ head_b,
                                                float* __restrict__ out,
                                                int nsamp) {
  __shared__ float Alds[2 * DIM * DIM];
  for (int i = threadIdx.x; i < 2 * DIM * DIM; i += blockDim.x) Alds[i] = ws[i];
  __syncthreads();

  const int lane = threadIdx.x & 31;
  const int wave = threadIdx.x >> 5;
  const int lo   = lane & 15;
  const bool hi  = (lane >> 4) != 0;     // false: lanes 0-15, true: lanes 16-31

  // A operands (A_j rows in WMMA f32 A-layout), loop-invariant:
  // chunk c: lane lo holds A_j[lo][4c+2*hi] / [4c+2*hi+1].
  v2f A0[4], A1[4];
#pragma unroll
  for (int c = 0; c < 4; ++c) {
    const int k = 4 * c + (hi ? 2 : 0);
    A0[c].x = Alds[lo * DIM + k];
    A0[c].y = Alds[lo * DIM + k + 1];
    A1[c].x = Alds[DIM * DIM + lo * DIM + k];
    A1[c].y = Alds[DIM * DIM + lo * DIM + k + 1];
  }
  const float hb0 = head_b[0], hb1 = head_b[1];

  const int ntiles = nsamp >> 4;
  const int waveId = blockIdx.x * 8 + wave;
  const int nwaves = gridDim.x * 8;

  for (int t = waveId; t < ntiles; t += nwaves) {
    const int base = t << 4;
    const int s    = base + lo;          // both halves own sample (base+lo)
    const float4 xv = ((const float4*)x)[s];

    float s0, c0, s1, c1, s2, c2, s3, c3;
    __sincosf(0.5f * xv.x, &s0, &c0);
    __sincosf(0.5f * xv.y, &s1, &c1);
    __sincosf(0.5f * xv.z, &s2, &c2);
    __sincosf(0.5f * xv.w, &s3, &c3);

    // rank-1 factors of psi: psi[n] = ab[n>>2] * cd[n&3]
    const float ab0 = c0 * c1, ab1 = c0 * s1, ab2 = s0 * c1, ab3 = s0 * s1;
    const float cd0 = c2 * c3, cd1 = c2 * s3, cd2 = s2 * c3, cd3 = s2 * s3;

    // half-dependent factor selection: 4 scalar cndmasks, no array indexing
    const float cdl0 = hi ? cd2 : cd0;   // cd[2*hi + 0]
    const float cdl1 = hi ? cd3 : cd1;   // cd[2*hi + 1]
    const float abl0 = hi ? ab2 : ab0;   // ab[0 + 2*hi]
    const float abl1 = hi ? ab3 : ab1;   // ab[1 + 2*hi]

    // W_j = A_j * Psi^T : B-operand chunk c is psi[4c+2hi+{0,1}] = ab_c*cdl{0,1}
    v8f acc0 = {}; v8f acc1 = {};
    {
      v2f b;
      b.x = ab0 * cdl0; b.y = ab0 * cdl1;
      acc0 = __builtin_amdgcn_wmma_f32_16x16x4_f32(false, A0[0], false, b,
                                                   (short)0, acc0, false, false);
      acc1 = __builtin_amdgcn_wmma_f32_16x16x4_f32(false, A1[0], false, b,
                                                   (short)0, acc1, false, false);
      b.x = ab1 * cdl0; b.y = ab1 * cdl1;
      acc0 = __builtin_amdgcn_wmma_f32_16x16x4_f32(false, A0[1], false, b,
                                                   (short)0, acc0, false, false);
      acc1 = __builtin_amdgcn_wmma_f32_16x16x4_f32(false, A1[1], false, b,
                                                   (short)0, acc1, false, false);
      b.x = ab2 * cdl0; b.y = ab2 * cdl1;
      acc0 = __builtin_amdgcn_wmma_f32_16x16x4_f32(false, A0[2], false, b,
                                                   (short)0, acc0, false, false);
      acc1 = __builtin_amdgcn_wmma_f32_16x16x4_f32(false, A1[2], false, b,
                                                   (short)0, acc1, false, false);
      b.x = ab3 * cdl0; b.y = ab3 * cdl1;
      acc0 = __builtin_amdgcn_wmma_f32_16x16x4_f32(false, A0[3], false, b,
                                                   (short)0, acc0, false, false);
      acc1 = __builtin_amdgcn_wmma_f32_16x16x4_f32(false, A1[3], false, b,
                                                   (short)0, acc1, false, false);
    }

    // Lane lo holds W_j[m + 8*hi][lo] = Y_j[sample lo][m + 8*hi] in acc[m].
    // Dot weights: psi[m + 8*hi] = abl_{m>>2} * cd[m&3]  (all-constant cd picks)
    const float p0 = abl0 * cd0, p1 = abl0 * cd1, p2 = abl0 * cd2, p3 = abl0 * cd3;
    const float p4 = abl1 * cd0, p5 = abl1 * cd1, p6 = abl1 * cd2, p7 = abl1 * cd3;

    float r0, r1;
    r0 = p0 * acc0[0];           r1 = p0 * acc1[0];
    r0 = fmaf(p1, acc0[1], r0);  r1 = fmaf(p1, acc1[1], r1);
    r0 = fmaf(p2, acc0[2], r0);  r1 = fmaf(p2, acc1[2], r1);
    r0 = fmaf(p3, acc0[3], r0);  r1 = fmaf(p3, acc1[3], r1);
    r0 = fmaf(p4, acc0[4], r0);  r1 = fmaf(p4, acc1[4], r1);
    r0 = fmaf(p5, acc0[5], r0);  r1 = fmaf(p5, acc1[5], r1);
    r0 = fmaf(p6, acc0[6], r0);  r1 = fmaf(p6, acc1[6], r1);
    r0 = fmaf(p7, acc0[7], r0);  r1 = fmaf(p7, acc1[7], r1);

    // combine the two halves (amplitudes 0-7 and 8-15) of the same sample
    r0 += __shfl_xor(r0, 16, 32);
    r1 += __shfl_xor(r1, 16, 32);

    if (!hi) {
      ((float2*)out)[base + lo] = make_float2(r0 + hb0, r1 + hb1);
    }
  }
}

// ---------------------------------------------------------------------------
extern "C" void kernel_launch(void* const* d_in, const int* in_sizes, int n_in,
                              void* d_out, int out_size, void* d_ws, size_t ws_size,
                              hipStream_t stream) {
  const float* x       = (const float*)d_in[0];
  const float* weights = (const float*)d_in[1];
  const float* head_w  = (const float*)d_in[2];
  const float* head_b  = (const float*)d_in[3];
  float* wsf = (float*)d_ws;
  const int nsamp = in_sizes[0] / 4;

  dvq_setup<<<1, 32, 0, stream>>>(weights, head_w, wsf);

  const int ntiles = nsamp >> 4;
  int blocks = (ntiles + 7) / 8;
  if (blocks > 1024) blocks = 1024;
  if (blocks < 1) blocks = 1;
  dvq_main<<<blocks, 256, 0, stream>>>(x, wsf, head_b, (float*)d_out, nsamp);
}